// SpatialNet_32495722561843
// MI455X (gfx1250) — compile-verified
//
#include <hip/hip_runtime.h>
#include <math.h>

// ---------------------------------------------------------------------------
// SpatialNet forward for MI455X (gfx1250), wave32 + WMMA fp32 path.
//   B=32, N=8, L=192, C=63, OFF=32 ; d_in=6160, d_h=6624, d_out=1536
// Stage 1: per-(b,n) interference features with LDS-resident kernel table.
// Stage 2: deterministic global sum-of-squares -> 1/norm.
// Stage 3: assemble x = [flat*inv, dss_min*inv, dss_max*inv, powers].
// Stage 4: three GEMMs on V_WMMA_F32_16X16X4_F32.
//          Memory-bound: 380MB of weights read exactly once -> 16us floor at
//          23.3 TB/s. K-split x4 (8 waves/block, 414 blocks) to get enough
//          bytes in flight to cover the HBM bandwidth-delay product;
//          nontemporal weight loads protect the hot activation rows in L2.
// ---------------------------------------------------------------------------

#define B_    32
#define N_    8
#define L_    192
#define C_    63
#define OFF_  32
#define NL_   (N_ * L_)              // 1536
#define FLAT_ (3 * NL_)              // 4608
#define DIN_  (FLAT_ + 2 * N_ + NL_) // 6160
#define DH_   (4 * NL_ + 2 * N_ * 30)// 6624
#define DOUT_ NL_                    // 1536

typedef __attribute__((ext_vector_type(2))) float v2f;
typedef __attribute__((ext_vector_type(8))) float v8f;

// ---------------------------------------------------------------- Stage 1 ---
__global__ __launch_bounds__(L_)
void feat_kernel(const int* __restrict__ T, const int* __restrict__ R,
                 const float* __restrict__ powers, const float* __restrict__ Kt,
                 float* __restrict__ flat, float* __restrict__ partials)
{
    __shared__ float kern[C_ * C_];
    __shared__ int   tx[L_], ty[L_], rx[L_], ry[L_];
    __shared__ float pw[L_];
    __shared__ float red[256];

    const int bn = blockIdx.x;           // b*N + n
    const int b  = bn / N_;
    const int n  = bn % N_;
    const int j  = threadIdx.x;

    for (int i = j; i < C_ * C_; i += L_) kern[i] = Kt[n * C_ * C_ + i];
    {
        const int* tp = T + ((size_t)bn * L_ + j) * 4;
        tx[j] = tp[2]; ty[j] = tp[3];
        const int* rp = R + ((size_t)bn * L_ + j) * 4;
        rx[j] = rp[2]; ry[j] = rp[3];
        pw[j] = powers[bn * L_ + j];
    }
    __syncthreads();

    const int txj = tx[j], tyj = ty[j], rxj = rx[j], ryj = ry[j];
    float sT = 0.f, sR = 0.f;
#pragma unroll 4
    for (int i = 0; i < L_; ++i) {
        const float p = pw[i];
        int gx = tx[i] - rxj + OFF_;
        int gy = ty[i] - ryj + OFF_;
        if ((unsigned)gx < (unsigned)C_ && (unsigned)gy < (unsigned)C_)
            sT += p * kern[gx * C_ + gy];
        int hx = rx[i] - txj + OFF_;
        int hy = ry[i] - tyj + OFF_;
        if ((unsigned)hx < (unsigned)C_ && (unsigned)hy < (unsigned)C_)
            sR += p * kern[hx * C_ + hy];
    }
    const float dss = kern[(rxj - txj + OFF_) * C_ + (ryj - tyj + OFF_)];

    float* fb = flat + (size_t)b * FLAT_;
    fb[0 * NL_ + n * L_ + j] = sT;
    fb[1 * NL_ + n * L_ + j] = sR;
    fb[2 * NL_ + n * L_ + j] = dss;

    red[j] = sT * sT + sR * sR + dss * dss;
    if (j < 256 - L_) red[L_ + j] = 0.f;
    __syncthreads();
    for (int off = 128; off > 0; off >>= 1) {
        if (j < off) red[j] += red[j + off];
        __syncthreads();
    }
    if (j == 0) partials[bn] = red[0];
}

// ---------------------------------------------------------------- Stage 2 ---
__global__ __launch_bounds__(256)
void norm_kernel(const float* __restrict__ partials, float* __restrict__ invn)
{
    __shared__ float red[256];
    const int t = threadIdx.x;
    red[t] = partials[t];                 // exactly B_*N_ == 256 partials
    __syncthreads();
    for (int off = 128; off > 0; off >>= 1) {
        if (t < off) red[t] += red[t + off];
        __syncthreads();
    }
    if (t == 0) invn[0] = 1.0f / fmaxf(sqrtf(red[0]), 1e-12f);
}

// ---------------------------------------------------------------- Stage 3 ---
__global__ __launch_bounds__(256)
void assemble_kernel(const float* __restrict__ flat, const float* __restrict__ powers,
                     const float* __restrict__ invn, float* __restrict__ x)
{
    __shared__ float smin[256], smax[256];
    const int b = blockIdx.x, t = threadIdx.x;
    const float inv = invn[0];
    const float* fb = flat + (size_t)b * FLAT_;
    float* xb = x + (size_t)b * DIN_;

    for (int i = t; i < FLAT_; i += 256) xb[i] = fb[i] * inv;

    const int n = t >> 5, gl = t & 31;
    float mn = 3.4e38f, mx = -3.4e38f;
    for (int jj = gl; jj < L_; jj += 32) {
        const float v = fb[2 * NL_ + n * L_ + jj];
        mn = fminf(mn, v);
        mx = fmaxf(mx, v);
    }
    smin[t] = mn; smax[t] = mx;
    __syncthreads();
    for (int off = 16; off > 0; off >>= 1) {
        if (gl < off) {
            smin[t] = fminf(smin[t], smin[t + off]);
            smax[t] = fmaxf(smax[t], smax[t + off]);
        }
        __syncthreads();
    }
    if (gl == 0) {
        xb[FLAT_ + n]      = smin[t] * inv;
        xb[FLAT_ + N_ + n] = smax[t] * inv;
    }
    for (int i = t; i < NL_; i += 256) xb[FLAT_ + 2 * N_ + i] = powers[(size_t)b * NL_ + i];
}

// ---------------------------------------------------------------- Stage 4 ---
// Block = 256 threads = 8 waves covering a 32(M) x 16(N) output tile with a
// 4-way K split: wave w -> m_tile = w&1, k_chunk = w>>1. Each wave runs
// V_WMMA_F32_16X16X4_F32 over K/4 with pointer-bumped operand loads:
//   A (16x4): lanes 0-15 hold K=0,1 of row M=lane; lanes 16-31 hold K=2,3
//             -> one b64 load per lane from row-major A (L2-resident).
//   B (4x16): VGPR v holds rows K=v, K=v+2 striped across lane halves
//             -> two nontemporal b32 loads per step from row-major W (stream).
// Partials combined through LDS in fixed order (deterministic), then
// bias + ReLU/sigmoid fused into the coalesced store.
template<int K, int NC, int ACT>
__global__ __launch_bounds__(256)
void gemm_kernel(const float* __restrict__ A, const float* __restrict__ W,
                 const float* __restrict__ bias, float* __restrict__ Out)
{
    constexpr int KC = K / 4;                 // K-range per wave
    __shared__ float red[8][8][32];           // [wave][v][lane] partials (8KB)

    const int lane  = threadIdx.x & 31;
    const int wave  = threadIdx.x >> 5;
    const int m_t   = wave & 1;               // 0/1 -> rows 0-15 / 16-31
    const int kc    = wave >> 1;              // 0..3 K-chunk
    const int m0    = m_t << 4;
    const int n0    = blockIdx.x * 16;
    const int l15   = lane & 15;
    const int khalf = (lane >> 4) << 1;       // 0 for lanes 0-15, 2 for 16-31

    const float* ap = A + (size_t)(m0 + l15) * K + kc * KC + khalf;
    const float* wp = W + (size_t)(kc * KC + khalf) * NC + n0 + l15;

    v8f acc = {};
#pragma unroll 8
    for (int k = 0; k < KC; k += 4) {
        v2f a = *(const v2f*)ap;                       // A[m][k..k+1]
        v2f bvec;
        bvec[0] = __builtin_nontemporal_load(wp);      // W[k+khalf  ][col]
        bvec[1] = __builtin_nontemporal_load(wp + NC); // W[k+khalf+1][col]
        acc = __builtin_amdgcn_wmma_f32_16x16x4_f32(
            /*neg_a=*/false, a, /*neg_b=*/false, bvec,
            /*c_mod=*/(short)0, acc, /*reuse_a=*/false, /*reuse_b=*/false);
        ap += 4;
        wp += 4 * (size_t)NC;
    }

    if (kc != 0) {
#pragma unroll
        for (int v = 0; v < 8; ++v) red[wave][v][lane] = acc[v];
    }
    __syncthreads();
    if (kc == 0) {
        const int col   = n0 + l15;
        const int rbase = m0 + ((lane >> 4) << 3);
        const float bv  = bias[col];
#pragma unroll
        for (int v = 0; v < 8; ++v) {
            // fixed combine order -> deterministic across calls
            float r = acc[v] + red[m_t + 2][v][lane]
                             + red[m_t + 4][v][lane]
                             + red[m_t + 6][v][lane];
            r += bv;
            r = (ACT == 0) ? fmaxf(r, 0.f) : (1.f / (1.f + __expf(-r)));
            Out[(size_t)(rbase + v) * NC + col] = r;
        }
    }
}

// ------------------------------------------------------------------ launch --
extern "C" void kernel_launch(void* const* d_in, const int* in_sizes, int n_in,
                              void* d_out, int out_size, void* d_ws, size_t ws_size,
                              hipStream_t stream)
{
    const int*   T      = (const int*)  d_in[0];
    const int*   R      = (const int*)  d_in[1];
    const float* powers = (const float*)d_in[2];
    const float* Kt     = (const float*)d_in[3];
    const float* W1     = (const float*)d_in[4];
    const float* b1     = (const float*)d_in[5];
    const float* W2     = (const float*)d_in[6];
    const float* b2     = (const float*)d_in[7];
    const float* W3     = (const float*)d_in[8];
    const float* b3     = (const float*)d_in[9];
    float*       out    = (float*)d_out;

    // workspace carve-up (~3.1 MB of floats)
    float* ws       = (float*)d_ws;
    float* flat     = ws;                    // B*FLAT  = 147456
    float* x        = flat + B_ * FLAT_;     // B*DIN   = 197120
    float* h1       = x    + B_ * DIN_;      // B*DH    = 211968
    float* h2       = h1   + B_ * DH_;       // B*DH    = 211968
    float* partials = h2   + B_ * DH_;       // 256
    float* invn     = partials + 256;        // 1

    feat_kernel    <<<B_ * N_, L_,  0, stream>>>(T, R, powers, Kt, flat, partials);
    norm_kernel    <<<1,       256, 0, stream>>>(partials, invn);
    assemble_kernel<<<B_,      256, 0, stream>>>(flat, powers, invn, x);

    gemm_kernel<DIN_, DH_,   0><<<DH_   / 16, 256, 0, stream>>>(x,  W1, b1, h1);
    gemm_kernel<DH_,  DH_,   0><<<DH_   / 16, 256, 0, stream>>>(h1, W2, b2, h2);
    gemm_kernel<DH_,  DOUT_, 1><<<DOUT_ / 16, 256, 0, stream>>>(h2, W3, b3, out);
}